// SiamFC_89369679495281
// MI455X (gfx1250) — compile-verified
//
#include <hip/hip_runtime.h>

typedef float v2f __attribute__((ext_vector_type(2)));
typedef float v8f __attribute__((ext_vector_type(8)));

#define C_DIM 512
#define P_DIM 1024    // 32*32 template positions
#define N_DIM 65536   // 256*256 search positions
#define B_DIM 3
#define OUT_SCALE 0.001f

// ---------------------------------------------------------------------------
// Phase 1: s[c] = OUT_SCALE * sum_p z[c, p]          (z is 512 x 1024 fp32)
// One block per channel; 256 threads each reduce a float4, then LDS tree.
// ---------------------------------------------------------------------------
__global__ __launch_bounds__(256)
void siamfc_reduce_z(const float* __restrict__ z, float* __restrict__ s) {
    __shared__ float red[256];
    const int c = blockIdx.x;
    const float4* zp = (const float4*)(z + (size_t)c * P_DIM);
    float4 v = zp[threadIdx.x];                 // 1024 floats = 256 float4
    red[threadIdx.x] = v.x + v.y + v.z + v.w;
    __syncthreads();
    for (int off = 128; off > 0; off >>= 1) {
        if (threadIdx.x < off) red[threadIdx.x] += red[threadIdx.x + off];
        __syncthreads();
    }
    if (threadIdx.x == 0) s[c] = red[0] * OUT_SCALE;
}

// ---------------------------------------------------------------------------
// Phase 2: out[b, n] = sum_c s[c] * x[b, c, n]   via V_WMMA_F32_16X16X4_F32
//
// Per wave: one 16-pixel tile of one batch.
//   A (16x4)  : s[c+k] replicated down all 16 M rows (broadcast from LDS)
//   B (4x16)  : x[b, c+k, n0+j]  (two coalesced 64B row segments per load)
//   D (16x16) : every row identical == out[n0 .. n0+15] after the c-loop
// Two independent accumulator chains hide WMMA latency.
// ---------------------------------------------------------------------------
__global__ __launch_bounds__(256)
void siamfc_corr_wmma(const float* __restrict__ x,
                      const float* __restrict__ s,
                      float* __restrict__ out) {
    __shared__ float sh_s[C_DIM];
    for (int i = threadIdx.x; i < C_DIM; i += 256) sh_s[i] = s[i];
    __syncthreads();

    const int lane   = threadIdx.x & 31;
    const int wave   = threadIdx.x >> 5;
    const int tile   = blockIdx.x * 8 + wave;   // 12288 tiles total
    const int b      = tile >> 12;              // tile / 4096
    const int n0     = (tile & 4095) << 4;      // 16 pixels per tile

    const int col    = lane & 15;               // N index within tile
    const int rowoff = (lane >> 4) << 1;        // K sub-row: 0 or 2

    // Per-lane base pointer into x for (b, c=rowoff, n0+col)
    const float* xp = x + (size_t)b * C_DIM * N_DIM
                        + (size_t)rowoff * N_DIM + (size_t)(n0 + col);
    const float* sp = sh_s + rowoff;

    v8f acc0 = {};
    v8f acc1 = {};

    #pragma unroll 4
    for (int c = 0; c < C_DIM; c += 8) {
        // ---- chain 0: channels c .. c+3 ----
        v2f a0, b0;
        a0.x = sp[c];
        a0.y = sp[c + 1];
        b0.x = xp[(size_t)c * N_DIM];
        b0.y = xp[(size_t)c * N_DIM + N_DIM];
        acc0 = __builtin_amdgcn_wmma_f32_16x16x4_f32(
                   false, a0, false, b0, (short)0, acc0, false, false);

        // ---- chain 1: channels c+4 .. c+7 ----
        v2f a1, b1;
        a1.x = sp[c + 4];
        a1.y = sp[c + 5];
        b1.x = xp[(size_t)(c + 4) * N_DIM];
        b1.y = xp[(size_t)(c + 4) * N_DIM + N_DIM];
        acc1 = __builtin_amdgcn_wmma_f32_16x16x4_f32(
                   false, a1, false, b1, (short)0, acc1, false, false);
    }

    // Row M=0 of D lives in acc[0] on lanes 0..15 (N = lane) -> 64B store.
    if (lane < 16) {
        out[(size_t)b * N_DIM + (size_t)(n0 + col)] = acc0[0] + acc1[0];
    }
}

// ---------------------------------------------------------------------------
extern "C" void kernel_launch(void* const* d_in, const int* in_sizes, int n_in,
                              void* d_out, int out_size, void* d_ws, size_t ws_size,
                              hipStream_t stream) {
    const float* z = (const float*)d_in[0];   // (1, 512, 32, 32)
    const float* x = (const float*)d_in[1];   // (3, 512, 256, 256)
    float* out = (float*)d_out;               // (3, 1, 256, 256)
    float* s   = (float*)d_ws;                // 512 floats of scratch

    siamfc_reduce_z<<<C_DIM, 256, 0, stream>>>(z, s);

    const int tiles  = B_DIM * (N_DIM / 16);  // 12288
    const int blocks = tiles / 8;             // 8 waves per block
    siamfc_corr_wmma<<<blocks, 256, 0, stream>>>(x, s, out);
}